// D3Net_31147102831224
// MI455X (gfx1250) — compile-verified
//
#include <hip/hip_runtime.h>
#include <cstdint>
#include <cstddef>

// ---------------------------------------------------------------------------
// D3Net forward for MI455X (gfx1250, wave32).
// All 1x1 convolutions (the dominant compute) use V_WMMA_F32_16X16X4_F32.
// Layout: NCHW fp32, B=32, C=128, H=W=64, HW=4096.
// ---------------------------------------------------------------------------

#define BN 32
#define HW 4096
#define HW_SHIFT 12
#define M_TOTAL (BN * HW) // 131072 (multiple of 16; tiles never straddle batches)

typedef float v2f __attribute__((ext_vector_type(2)));
typedef float v8f __attribute__((ext_vector_type(8)));

// ---------------------------------------------------------------------------
// FP32 WMMA GEMM for 1x1 conv:
//   out[n,o,hw] = f( sum_k W[o,k]*in[n,k,hw] + bias[o] (+ Dadd[n,Doff+o,hw]) )
//
// Role assignment (chosen for coalescing):
//   A (16x4)  = weight tile : M = 16 output channels, K chunk of 4.
//               Row-contiguous -> one 8-byte load per lane.
//   B (4x16)  = input tile  : N = 16 consecutive spatial positions.
//   C (16x16) : acc[j] = (channel o0 + j + 8*half, hw hw0 + l15)
//               -> stores/residual-loads are 2x64B coalesced segments.
//
// VGPR fragment layouts per CDNA5 ISA 7.12.2:
//   A: lanes 0-15 & 16-31 hold M=0..15; VGPR0 = K(=2*half), VGPR1 = K+1
//   B: VGPR0 = row K=2*half (N = lane&15), VGPR1 = row K+1
//   C: acc[j] = row (j + 8*half), col (lane&15)
//
// The optional channel-concat input (inA: CinA ch, inA2: Cin-CinA ch) is
// handled with two wave-uniform K loops (CinA is a multiple of 4), so the
// hot loop has no divergent control flow.
// ---------------------------------------------------------------------------
__global__ __launch_bounds__(256) void gemm1x1_wmma(
    const float* __restrict__ inA, const float* __restrict__ inA2,
    const float* __restrict__ W, const float* __restrict__ bias,
    const float* __restrict__ Dadd, float* __restrict__ out,
    int Cin, int CinA, int Cout, int Dcin, int Doff, int leaky)
{
  const int nst   = M_TOTAL >> 4;          // spatial tiles (fastest)
  const int tiles = nst * (Cout >> 4);
  const int wid   = blockIdx.x * (blockDim.x >> 5) + (threadIdx.x >> 5);
  const int nwv   = gridDim.x * (blockDim.x >> 5);
  const int lane  = threadIdx.x & 31;
  const int half  = lane >> 4;
  const int l15   = lane & 15;
  const int kofs  = half << 1;             // K offset within chunk for this half

  for (int t = wid; t < tiles; t += nwv) {
    const int st  = t % nst;               // spatial tile
    const int ot  = t / nst;               // out-channel tile
    const int m0  = st << 4;
    const int o0  = ot << 4;
    const int nb  = m0 >> HW_SHIFT;        // batch (tiles stay inside a batch)
    const int hw0 = m0 & (HW - 1);
    const int hwl = hw0 + l15;             // this lane's spatial column

    // Weight row for A fragment: lane l15 -> output channel o0+l15.
    const float* wbase = W + (size_t)(o0 + l15) * Cin + kofs;
    v8f acc = {};

    // ---- K loop over first (or only) input tensor ----
    const float* bsrc = inA + ((size_t)nb * CinA + kofs) * HW + hwl;
    for (int k0 = 0; k0 < CinA; k0 += 4) {
      const v2f a = *(const v2f*)(wbase + k0);     // W[o][k], W[o][k+1]
      v2f b;
      b.x = bsrc[0];                               // in[ch k][hw]
      b.y = bsrc[HW];                              // in[ch k+1][hw]
      bsrc += 4 * HW;
      acc = __builtin_amdgcn_wmma_f32_16x16x4_f32(
          false, a, false, b, (short)0, acc, false, false);
    }
    // ---- K loop over second concat tensor (wave-uniform trip count) ----
    if (CinA < Cin) {
      const float* bsrc2 = inA2 + ((size_t)nb * (Cin - CinA) + kofs) * HW + hwl;
      for (int k0 = CinA; k0 < Cin; k0 += 4) {
        const v2f a = *(const v2f*)(wbase + k0);
        v2f b;
        b.x = bsrc2[0];
        b.y = bsrc2[HW];
        bsrc2 += 4 * HW;
        acc = __builtin_amdgcn_wmma_f32_16x16x4_f32(
            false, a, false, b, (short)0, acc, false, false);
      }
    }

    // Epilogue: bias (+ residual tensor) (+ leaky relu); coalesced stores
    // (per j: lanes 0-15 and 16-31 each cover 16 consecutive hw of a channel).
#pragma unroll
    for (int j = 0; j < 8; ++j) {
      const int o = o0 + j + (half << 3);
      float v = acc[j] + bias[o];
      if (Dadd) v += Dadd[((size_t)nb * Dcin + Doff + o) * HW + hwl];
      if (leaky) v = v > 0.0f ? v : 0.01f * v;
      out[((size_t)nb * Cout + o) * HW + hwl] = v;
    }
  }
}

// ---------------------------------------------------------------------------
// Depthwise 7x7 'SAME' conv over 248 channels; input = fused channels 8..255.
// ---------------------------------------------------------------------------
__global__ void dwconv7x7(const float* __restrict__ fused,
                          const float* __restrict__ W,
                          const float* __restrict__ bias,
                          float* __restrict__ out)
{
  const int idx = blockIdx.x * blockDim.x + threadIdx.x;
  const int total = BN * 248 * HW;
  if (idx >= total) return;
  const int w = idx & 63;
  const int h = (idx >> 6) & 63;
  const int c = (idx >> HW_SHIFT) % 248;
  const int n = idx / (248 * HW);
  const float* src = fused + ((size_t)n * 256 + (c + 8)) * HW;
  const float* wt  = W + c * 49;
  float s = bias[c];
#pragma unroll
  for (int dy = 0; dy < 7; ++dy) {
    const int y = h + dy - 3;
    if (y < 0 || y > 63) continue;
#pragma unroll
    for (int dx = 0; dx < 7; ++dx) {
      const int x = w + dx - 3;
      if (x < 0 || x > 63) continue;
      s += wt[dy * 7 + dx] * src[y * 64 + x];
    }
  }
  out[idx] = s;
}

// ---------------------------------------------------------------------------
// Decision net: conv3x3 'SAME' + ReLU + 2x2 maxpool, one output per thread.
// ---------------------------------------------------------------------------
__global__ void conv3_relu_pool(const float* __restrict__ in,
                                const float* __restrict__ W,
                                const float* __restrict__ bias,
                                float* __restrict__ out,
                                int Cin, int Cout, int Hin)
{
  const int Hout = Hin >> 1;
  const int total = BN * Cout * Hout * Hout;
  const int idx = blockIdx.x * blockDim.x + threadIdx.x;
  if (idx >= total) return;
  const int wo = idx % Hout;
  const int ho = (idx / Hout) % Hout;
  const int o  = (idx / (Hout * Hout)) % Cout;
  const int n  = idx / (Cout * Hout * Hout);
  float best = -3.4e38f;
  for (int py = 0; py < 2; ++py)
    for (int px = 0; px < 2; ++px) {
      const int h = ho * 2 + py, w = wo * 2 + px;
      float s = bias[o];
      for (int i = 0; i < Cin; ++i) {
        const float* src = in + ((size_t)n * Cin + i) * Hin * Hin;
        const float* wt  = W + ((size_t)o * Cin + i) * 9;
#pragma unroll
        for (int dy = 0; dy < 3; ++dy) {
          const int y = h + dy - 1;
          if (y < 0 || y >= Hin) continue;
#pragma unroll
          for (int dx = 0; dx < 3; ++dx) {
            const int x = w + dx - 1;
            if (x < 0 || x >= Hin) continue;
            s += wt[dy * 3 + dx] * src[y * Hin + x];
          }
        }
      }
      s = fmaxf(s, 0.0f);       // ReLU, then 2x2 max pool
      best = fmaxf(best, s);
    }
  out[idx] = best;
}

// Hash-based uniform in (1e-6, 1-1e-6). NOTE: jax threefry(key=7) is not
// reproducible on-device; this provides a deterministic stand-in Gumbel draw.
__device__ __forceinline__ float hash_u01(unsigned v) {
  v ^= v >> 17; v *= 0xed5ad4bbu; v ^= v >> 11; v *= 0xac4c1b51u;
  v ^= v >> 15; v *= 0x31848babu; v ^= v >> 14;
  float u = (float)(v >> 8) * (1.0f / 16777216.0f);
  return fminf(fmaxf(u, 1e-6f), 1.0f - 1e-6f);
}

// mean over 4x4 -> 1x1 conv (8->2) -> softmax -> +gumbel -> per-batch decision
__global__ void dn_head(const float* __restrict__ s4 /*(B,8,4,4)*/,
                        const float* __restrict__ W5 /*(2,8)*/,
                        const float* __restrict__ b5 /*(2)*/,
                        float* __restrict__ dec)
{
  const int b = threadIdx.x;
  if (b >= BN) return;
  float f[8];
#pragma unroll
  for (int j = 0; j < 8; ++j) {
    const float* p = s4 + ((size_t)b * 8 + j) * 16;
    float s = 0.0f;
#pragma unroll
    for (int t = 0; t < 16; ++t) s += p[t];
    f[j] = s * (1.0f / 16.0f);
  }
  float l0 = b5[0], l1 = b5[1];
#pragma unroll
  for (int j = 0; j < 8; ++j) { l0 += W5[j] * f[j]; l1 += W5[8 + j] * f[j]; }
  const float mx = fmaxf(l0, l1);
  const float e0 = expf(l0 - mx), e1 = expf(l1 - mx);
  const float fc0 = e0 / (e0 + e1), fc1 = e1 / (e0 + e1);
  const float g0 = -logf(-logf(hash_u01(0x9E3779B9u * (7u + 2u * b + 0u))));
  const float g1 = -logf(-logf(hash_u01(0x9E3779B9u * (7u + 2u * b + 1u))));
  dec[b] = (fc1 + g1 > fc0 + g0) ? 1.0f : 0.0f;
}

// ---------------------------------------------------------------------------
// LayerNorm over channel dim (C=128), one pixel per thread (coalesced per c).
// ---------------------------------------------------------------------------
__global__ void ln2d(const float* __restrict__ in, const float* __restrict__ g,
                     const float* __restrict__ bb, float* __restrict__ out)
{
  const int p = blockIdx.x * blockDim.x + threadIdx.x;
  if (p >= BN * HW) return;
  const int n = p >> HW_SHIFT, hw = p & (HW - 1);
  const float* base = in + ((size_t)n << 19) + hw;   // n*128*HW
  float s = 0.0f, ss = 0.0f;
  for (int c = 0; c < 128; ++c) {
    const float v = base[c << HW_SHIFT];
    s += v; ss += v * v;
  }
  const float mu = s * (1.0f / 128.0f);
  const float var = ss * (1.0f / 128.0f) - mu * mu;
  const float rs = rsqrtf(var + 1e-6f);
  float* ob = out + ((size_t)n << 19) + hw;
  for (int c = 0; c < 128; ++c) {
    const float v = base[c << HW_SHIFT];
    ob[c << HW_SHIFT] = (v - mu) * rs * g[c] + bb[c];
  }
}

// ---------------------------------------------------------------------------
// Elementwise kernels
// ---------------------------------------------------------------------------
__global__ void add2(const float* __restrict__ a, const float* __restrict__ b,
                     float* __restrict__ o, int n)
{
  const int i = blockIdx.x * blockDim.x + threadIdx.x;
  if (i < n) o[i] = a[i] + b[i];
}

// y0 = pwa * dw0: fused channels 0..7 times dw channels 0..7 -> 8-channel out
__global__ void mul_gate8(const float* __restrict__ fused,
                          const float* __restrict__ dw, float* __restrict__ o)
{
  const int idx = blockIdx.x * blockDim.x + threadIdx.x;
  const int total = BN * 8 * HW;
  if (idx >= total) return;
  const int hw = idx & (HW - 1);
  const int c  = (idx >> HW_SHIFT) & 7;
  const int n  = idx >> 15;                       // /(8*HW)
  const float a = fused[((size_t)n * 256 + c) * HW + hw];
  const float b = dw[((size_t)n * 248 + c) * HW + hw];
  o[idx] = a * b;
}

// yres = lnx + h * beta[c]
__global__ void axpb(const float* __restrict__ lnx, const float* __restrict__ h,
                     const float* __restrict__ beta, float* __restrict__ o)
{
  const int idx = blockIdx.x * blockDim.x + threadIdx.x;
  const int total = BN * 128 * HW;
  if (idx >= total) return;
  const int c = (idx >> HW_SHIFT) & 127;
  o[idx] = lnx[idx] + h[idx] * beta[c];
}

// out = dec[n] ? (yres + h2*gamma[c])*res_scale + x : x
__global__ void final_combine(const float* __restrict__ x,
                              const float* __restrict__ yres,
                              const float* __restrict__ h2,
                              const float* __restrict__ gamma,
                              const float* __restrict__ res_scale,
                              const float* __restrict__ dec,
                              float* __restrict__ out)
{
  const int idx = blockIdx.x * blockDim.x + threadIdx.x;
  const int total = BN * 128 * HW;
  if (idx >= total) return;
  const int c = (idx >> HW_SHIFT) & 127;
  const int n = idx >> 19;                        // /(128*HW)
  const float xv = x[idx];
  const float v = (yres[idx] + h2[idx] * gamma[c]) * res_scale[0] + xv;
  out[idx] = dec[n] > 0.5f ? v : xv;
}

// ---------------------------------------------------------------------------
// Host orchestration
// ---------------------------------------------------------------------------
extern "C" void kernel_launch(void* const* d_in, const int* in_sizes, int n_in,
                              void* d_out, int out_size, void* d_ws, size_t ws_size,
                              hipStream_t stream)
{
  // Inputs in setup_inputs() dict order (recursive insertion order):
  const float* x      = (const float*)d_in[0];
  const float* cPromt = (const float*)d_in[1];
  const float* rPromt = (const float*)d_in[2];
  const float* res_sc = (const float*)d_in[3];
  // dn: 5 (w,b) pairs
  const float* dnw[5], *dnb[5];
  for (int i = 0; i < 5; ++i) { dnw[i] = (const float*)d_in[4 + 2*i]; dnb[i] = (const float*)d_in[5 + 2*i]; }
  const float* ln1g = (const float*)d_in[14]; const float* ln1b = (const float*)d_in[15];
  const float* ln2g = (const float*)d_in[16]; const float* ln2b = (const float*)d_in[17];
  const float* beta  = (const float*)d_in[18];
  const float* gamma = (const float*)d_in[19];
  const float* c11w  = (const float*)d_in[20]; const float* c11b = (const float*)d_in[21];
  // gnconv param block: proj_in(w,b), dwconv(w,b), pws[4](w,b), proj_out(w,b) -> 14 tensors
  struct GN { const float *piw, *pib, *dww, *dwb, *pw[4], *pb[4], *pow_, *pob; };
  auto loadGN = [&](int base) {
    GN g;
    g.piw = (const float*)d_in[base + 0];  g.pib = (const float*)d_in[base + 1];
    g.dww = (const float*)d_in[base + 2];  g.dwb = (const float*)d_in[base + 3];
    for (int i = 0; i < 4; ++i) { g.pw[i] = (const float*)d_in[base + 4 + 2*i]; g.pb[i] = (const float*)d_in[base + 5 + 2*i]; }
    g.pow_ = (const float*)d_in[base + 12]; g.pob = (const float*)d_in[base + 13];
    return g;
  };
  GN b1 = loadGN(22);
  GN b2 = loadGN(36);
  float* out = (float*)d_out;

  // Workspace carve (floats): FUSED(256ch) DW(248ch) P0..P3(128ch) DEC
  float* ws = (float*)d_ws;
  size_t off = 0;
  auto carve = [&](size_t nf) { float* p = ws + off; off += nf; return p; };
  float* FUSED = carve((size_t)BN * 256 * HW);
  float* DW    = carve((size_t)BN * 248 * HW);
  float* P0    = carve((size_t)BN * 128 * HW);
  float* P1    = carve((size_t)BN * 128 * HW);
  float* P2    = carve((size_t)BN * 128 * HW);
  float* P3    = carve((size_t)BN * 128 * HW);
  float* DEC   = carve(64);

  const int T128 = BN * 128 * HW;
  auto eblk = [](int n) { return (n + 255) / 256; };

  auto gemm = [&](const float* A, const float* A2, const float* Wt, const float* bs,
                  const float* D, float* O, int Cin, int CinA, int Cout,
                  int Dcin, int Doff, int leaky) {
    const int tiles  = (M_TOTAL >> 4) * (Cout >> 4);
    const int blocks = (tiles + 7) / 8;            // 8 waves / 256-thread block
    gemm1x1_wmma<<<blocks, 256, 0, stream>>>(A, A2, Wt, bs, D, O,
                                             Cin, CinA, Cout, Dcin, Doff, leaky);
  };

  // gnconv: FUSED = proj_in(concat(xin, cPromt)); DW = dwconv; gated pws chain.
  // Uses scratch bufA/bufB (ping-pong); writes leaky(proj_out(...)) into res.
  auto gnconv = [&](const GN& g, const float* xin, float* bufA, float* bufB, float* res) {
    gemm(xin, cPromt, g.piw, g.pib, nullptr, FUSED, 256, 128, 256, 0, 0, 0);
    dwconv7x7<<<eblk(BN * 248 * HW), 256, 0, stream>>>(FUSED, g.dww, g.dwb, DW);
    mul_gate8<<<eblk(BN * 8 * HW), 256, 0, stream>>>(FUSED, DW, bufA);
    gemm(bufA, nullptr, g.pw[0], g.pb[0], DW, bufB,   8,   8,  16, 248,   8, 0);
    gemm(bufB, nullptr, g.pw[1], g.pb[1], DW, bufA,  16,  16,  32, 248,  24, 0);
    gemm(bufA, nullptr, g.pw[2], g.pb[2], DW, bufB,  32,  32,  64, 248,  56, 0);
    gemm(bufB, nullptr, g.pw[3], g.pb[3], DW, bufA,  64,  64, 128, 248, 120, 0);
    gemm(bufA, nullptr, g.pow_, g.pob, nullptr, res, 128, 128, 128, 0, 0, /*leaky*/1);
  };

  // ---- Decision network (independent; uses P0/P1 before ADB needs them) ----
  add2<<<eblk(T128), 256, 0, stream>>>(x, rPromt, P0, T128);
  conv3_relu_pool<<<eblk(BN*64*32*32), 256, 0, stream>>>(P0, dnw[0], dnb[0], P1, 128, 64, 64);
  conv3_relu_pool<<<eblk(BN*32*16*16), 256, 0, stream>>>(P1, dnw[1], dnb[1], P0,  64, 32, 32);
  conv3_relu_pool<<<eblk(BN*16* 8* 8), 256, 0, stream>>>(P0, dnw[2], dnb[2], P1,  32, 16, 16);
  conv3_relu_pool<<<eblk(BN* 8* 4* 4), 256, 0, stream>>>(P1, dnw[3], dnb[3], P0,  16,  8,  8);
  dn_head<<<1, 32, 0, stream>>>(P0, dnw[4], dnb[4], DEC);

  // ---- ADB branch ----
  ln2d<<<eblk(BN * HW), 256, 0, stream>>>(x, ln1g, ln1b, P0);      // P0 = LN1(x)
  gnconv(b1, P0, P1, P2, P2);                                      // P2 = h = leaky(gnconv1)
  axpb<<<eblk(T128), 256, 0, stream>>>(P0, P2, beta, P3);          // P3 = y = lnx + h*beta
  ln2d<<<eblk(BN * HW), 256, 0, stream>>>(P2, ln2g, ln2b, P0);     // P0 = LN2(h)
  gemm(P0, nullptr, c11w, c11b, nullptr, P1, 128, 128, 128, 0, 0, 0); // P1 = conv11
  gnconv(b2, P1, P0, P2, P1);                                      // P1 = h2 = leaky(gnconv2)
  final_combine<<<eblk(T128), 256, 0, stream>>>(x, P3, P1, gamma, res_sc, DEC, out);
}